// ENTITY_DECODER_82798379533062
// MI455X (gfx1250) — compile-verified
//
#include <hip/hip_runtime.h>
#include <hip/hip_bf16.h>
#include <math.h>

// Problem constants (match reference)
constexpr int B  = 64;
constexpr int T  = 1024;
constexpr int CH = 256;
constexpr int V  = 32000;
constexpr int E  = 64;
constexpr int L  = 128;
constexpr int L4 = 4 * L;

constexpr int KSPLIT  = 64;            // split-K for the V=32000 reduction GEMM
constexpr int KCHUNK  = V / KSPLIT;    // 500 (multiple of 4)
constexpr int TCHUNKS = 8;             // T split for attention
constexpr int TCHUNK  = T / TCHUNKS;   // 128

typedef __attribute__((ext_vector_type(2))) float v2f;
typedef __attribute__((ext_vector_type(8))) float v8f;

__device__ __forceinline__ float sigmoidf_(float x) {
    return 1.0f / (1.0f + __expf(-x));
}

__device__ __forceinline__ v8f wmma_f32(v2f a, v2f b, v8f c) {
    return __builtin_amdgcn_wmma_f32_16x16x4_f32(
        false, a, false, b, (short)0, c, false, false);
}

// ---------------------------------------------------------------------------
// Kernel 1: lstm_inp[B,E] = prev_char@Wc (split-K, atomic accumulate)
//                          + prev_cont@Wct1 (done by ksplit==0 blocks)
// One wave owns the full M=64 strip of one 16-wide N column for its K chunk:
// B fragments (Wc) are loaded once per k-step and reused by 4 WMMAs.
// grid = (E/16 = 4, KSPLIT), block = 32.
// ---------------------------------------------------------------------------
__global__ __launch_bounds__(32) void k_lstm_inp(
    const float* __restrict__ prev_char, const float* __restrict__ Wc,
    const float* __restrict__ prev_cont, const float* __restrict__ Wct1,
    float* __restrict__ out /* ws: [B,E], pre-zeroed */) {
    const int n0   = blockIdx.x * 16;
    const int lane = threadIdx.x;
    const int ml   = lane & 15;          // row within a 16-row tile
    const int n    = n0 + ml;
    const int kh   = (lane >> 4) * 2;    // 0 or 2 (K sub-offset per lane half)

    v8f acc0 = {}, acc1 = {}, acc2 = {}, acc3 = {};

    const int kbeg = blockIdx.y * KCHUNK;
    for (int k = kbeg; k < kbeg + KCHUNK; k += 4) {
        v2f bf;                                       // shared B fragment
        bf.x = Wc[(k + kh + 0) * E + n];
        bf.y = Wc[(k + kh + 1) * E + n];
        v2f a0, a1, a2, a3;
        a0.x = prev_char[(ml +  0) * V + k + kh + 0];
        a0.y = prev_char[(ml +  0) * V + k + kh + 1];
        a1.x = prev_char[(ml + 16) * V + k + kh + 0];
        a1.y = prev_char[(ml + 16) * V + k + kh + 1];
        a2.x = prev_char[(ml + 32) * V + k + kh + 0];
        a2.y = prev_char[(ml + 32) * V + k + kh + 1];
        a3.x = prev_char[(ml + 48) * V + k + kh + 0];
        a3.y = prev_char[(ml + 48) * V + k + kh + 1];
        acc0 = wmma_f32(a0, bf, acc0);
        acc1 = wmma_f32(a1, bf, acc1);
        acc2 = wmma_f32(a2, bf, acc2);
        acc3 = wmma_f32(a3, bf, acc3);
    }

    // prev_cont @ Wct1 (K = CH = 256), only on split 0
    if (blockIdx.y == 0) {
        for (int k = 0; k < CH; k += 4) {
            v2f bf;
            bf.x = Wct1[(k + kh + 0) * E + n];
            bf.y = Wct1[(k + kh + 1) * E + n];
            v2f a0, a1, a2, a3;
            a0.x = prev_cont[(ml +  0) * CH + k + kh + 0];
            a0.y = prev_cont[(ml +  0) * CH + k + kh + 1];
            a1.x = prev_cont[(ml + 16) * CH + k + kh + 0];
            a1.y = prev_cont[(ml + 16) * CH + k + kh + 1];
            a2.x = prev_cont[(ml + 32) * CH + k + kh + 0];
            a2.y = prev_cont[(ml + 32) * CH + k + kh + 1];
            a3.x = prev_cont[(ml + 48) * CH + k + kh + 0];
            a3.y = prev_cont[(ml + 48) * CH + k + kh + 1];
            acc0 = wmma_f32(a0, bf, acc0);
            acc1 = wmma_f32(a1, bf, acc1);
            acc2 = wmma_f32(a2, bf, acc2);
            acc3 = wmma_f32(a3, bf, acc3);
        }
    }

    const int rbase = (lane >> 4) * 8;   // C/D layout: upper lane-half holds M+8
    #pragma unroll
    for (int r = 0; r < 8; ++r) {
        atomicAdd(&out[( 0 + r + rbase) * E + n], acc0[r]);
        atomicAdd(&out[(16 + r + rbase) * E + n], acc1[r]);
        atomicAdd(&out[(32 + r + rbase) * E + n], acc2[r]);
        atomicAdd(&out[(48 + r + rbase) * E + n], acc3[r]);
    }
}

// ---------------------------------------------------------------------------
// Kernel 2: LSTM cell. grid = B, block = L.
// ---------------------------------------------------------------------------
__global__ __launch_bounds__(L) void k_lstm_cell(
    const float* __restrict__ lstm_inp, const float* __restrict__ state_h,
    const float* __restrict__ state_c,
    const float* __restrict__ Wk /* [E,4L] */,
    const float* __restrict__ Wr /* [L,4L] */,
    float* __restrict__ out_h, float* __restrict__ out_c) {
    __shared__ float sx[E];
    __shared__ float sh[L];
    const int b = blockIdx.x, l = threadIdx.x;
    if (l < E) sx[l] = lstm_inp[b * E + l];
    sh[l] = state_h[b * L + l];
    __syncthreads();

    float z0 = 0.f, z1 = 0.f, z2 = 0.f, z3 = 0.f;
    for (int e = 0; e < E; ++e) {
        const float x = sx[e];
        const float* row = Wk + e * L4;
        z0 = fmaf(x, row[l], z0);
        z1 = fmaf(x, row[L + l], z1);
        z2 = fmaf(x, row[2 * L + l], z2);
        z3 = fmaf(x, row[3 * L + l], z3);
    }
    for (int h = 0; h < L; ++h) {
        const float x = sh[h];
        const float* row = Wr + h * L4;
        z0 = fmaf(x, row[l], z0);
        z1 = fmaf(x, row[L + l], z1);
        z2 = fmaf(x, row[2 * L + l], z2);
        z3 = fmaf(x, row[3 * L + l], z3);
    }
    const float i = sigmoidf_(z0);
    const float f = sigmoidf_(z1);
    const float g = tanhf(z2);
    const float o = sigmoidf_(z3);
    const float c = f * state_c[b * L + l] + i * g;
    out_h[b * L + l] = o * tanhf(c);
    out_c[b * L + l] = fminf(10.f, fmaxf(-10.f, c));
}

// ---------------------------------------------------------------------------
// Kernel 3a: attention partial sums. grid = (B, TCHUNKS), block = CH.
// Single streaming pass over F (64 MB -> dominant HBM term).
// exp without max-subtraction is safe: |e| <= |V_attn[ch]| (small).
// ---------------------------------------------------------------------------
__global__ __launch_bounds__(CH) void k_attn_partial(
    const float* __restrict__ new_h, const float* __restrict__ W_h,
    const float* __restrict__ V_attn, const float* __restrict__ F,
    float* __restrict__ num, float* __restrict__ den) {
    const int b = blockIdx.x, ch = threadIdx.x;
    __shared__ float sh[L];
    if (ch < L) sh[ch] = new_h[b * L + ch];
    __syncthreads();

    float ht = 0.f;
    for (int l = 0; l < L; ++l) ht = fmaf(sh[l], W_h[l * CH + ch], ht);
    const float va = V_attn[ch];

    const int t0 = blockIdx.y * TCHUNK;
    const float* Fp = F + (size_t)b * T * CH + (size_t)t0 * CH + ch;
    float ln = 0.f, ld = 0.f;
    for (int t = 0; t < TCHUNK; ++t) {
        if (t + 16 < TCHUNK)
            __builtin_prefetch(Fp + (t + 16) * CH, 0, 0);   // global_prefetch_b8
        const float fv = Fp[t * CH];                        // coalesced across ch
        const float w = __expf(va * tanhf(ht + fv));
        ld += w;
        ln = fmaf(w, fv, ln);
    }
    atomicAdd(&num[b * CH + ch], ln);
    atomicAdd(&den[b * CH + ch], ld);
}

// Kernel 3b: c_t = num/den. grid = B, block = CH.
__global__ __launch_bounds__(CH) void k_attn_final(
    const float* __restrict__ num, const float* __restrict__ den,
    float* __restrict__ c_t) {
    const int idx = blockIdx.x * CH + threadIdx.x;
    c_t[idx] = num[idx] / den[idx];
}

// ---------------------------------------------------------------------------
// Kernel 4: logits[B,V] = new_h@Wo + c_t@Wct2 (f32 WMMA).
// One wave owns the full M=64 strip of a 16-wide N column: each B fragment
// (Wo/Wct2, the 49 MB term) is loaded exactly once and reused by 4 WMMAs.
// grid = V/16 = 2000, block = 32.
// ---------------------------------------------------------------------------
__global__ __launch_bounds__(32) void k_logits(
    const float* __restrict__ new_h, const float* __restrict__ Wo,
    const float* __restrict__ c_t, const float* __restrict__ Wct2,
    float* __restrict__ logits) {
    const int n0   = blockIdx.x * 16;
    const int lane = threadIdx.x;
    const int ml   = lane & 15;
    const int n    = n0 + ml;
    const int kh   = (lane >> 4) * 2;

    v8f acc0 = {}, acc1 = {}, acc2 = {}, acc3 = {};

    for (int k = 0; k < L; k += 4) {           // new_h @ Wo, K = 128
        v2f bf;
        bf.x = Wo[(k + kh + 0) * V + n];       // coalesced across n
        bf.y = Wo[(k + kh + 1) * V + n];
        v2f a0, a1, a2, a3;
        a0.x = new_h[(ml +  0) * L + k + kh + 0];
        a0.y = new_h[(ml +  0) * L + k + kh + 1];
        a1.x = new_h[(ml + 16) * L + k + kh + 0];
        a1.y = new_h[(ml + 16) * L + k + kh + 1];
        a2.x = new_h[(ml + 32) * L + k + kh + 0];
        a2.y = new_h[(ml + 32) * L + k + kh + 1];
        a3.x = new_h[(ml + 48) * L + k + kh + 0];
        a3.y = new_h[(ml + 48) * L + k + kh + 1];
        acc0 = wmma_f32(a0, bf, acc0);
        acc1 = wmma_f32(a1, bf, acc1);
        acc2 = wmma_f32(a2, bf, acc2);
        acc3 = wmma_f32(a3, bf, acc3);
    }
    for (int k = 0; k < CH; k += 4) {          // c_t @ Wct2, K = 256
        v2f bf;
        bf.x = Wct2[(k + kh + 0) * V + n];
        bf.y = Wct2[(k + kh + 1) * V + n];
        v2f a0, a1, a2, a3;
        a0.x = c_t[(ml +  0) * CH + k + kh + 0];
        a0.y = c_t[(ml +  0) * CH + k + kh + 1];
        a1.x = c_t[(ml + 16) * CH + k + kh + 0];
        a1.y = c_t[(ml + 16) * CH + k + kh + 1];
        a2.x = c_t[(ml + 32) * CH + k + kh + 0];
        a2.y = c_t[(ml + 32) * CH + k + kh + 1];
        a3.x = c_t[(ml + 48) * CH + k + kh + 0];
        a3.y = c_t[(ml + 48) * CH + k + kh + 1];
        acc0 = wmma_f32(a0, bf, acc0);
        acc1 = wmma_f32(a1, bf, acc1);
        acc2 = wmma_f32(a2, bf, acc2);
        acc3 = wmma_f32(a3, bf, acc3);
    }

    const int rbase = (lane >> 4) * 8;
    #pragma unroll
    for (int r = 0; r < 8; ++r) {
        logits[(size_t)( 0 + r + rbase) * V + n] = acc0[r];
        logits[(size_t)(16 + r + rbase) * V + n] = acc1[r];
        logits[(size_t)(32 + r + rbase) * V + n] = acc2[r];
        logits[(size_t)(48 + r + rbase) * V + n] = acc3[r];
    }
}

// ---------------------------------------------------------------------------
// Kernel 5: row softmax over V, in place. grid = B, block = 512.
// ---------------------------------------------------------------------------
__global__ __launch_bounds__(512) void k_softmax_rows(float* __restrict__ logits) {
    __shared__ float red[512];
    const int b = blockIdx.x, tid = threadIdx.x;
    float* row = logits + (size_t)b * V;

    float m = -3.4e38f;
    for (int i = tid; i < V; i += 512) m = fmaxf(m, row[i]);
    red[tid] = m;
    __syncthreads();
    for (int s = 256; s > 0; s >>= 1) {
        if (tid < s) red[tid] = fmaxf(red[tid], red[tid + s]);
        __syncthreads();
    }
    m = red[0];
    __syncthreads();

    float sum = 0.f;
    for (int i = tid; i < V; i += 512) {
        const float e = __expf(row[i] - m);
        row[i] = e;
        sum += e;
    }
    red[tid] = sum;
    __syncthreads();
    for (int s = 256; s > 0; s >>= 1) {
        if (tid < s) red[tid] += red[tid + s];
        __syncthreads();
    }
    const float inv = 1.0f / red[0];
    __syncthreads();
    for (int i = tid; i < V; i += 512) row[i] *= inv;
}

// ---------------------------------------------------------------------------
extern "C" void kernel_launch(void* const* d_in, const int* in_sizes, int n_in,
                              void* d_out, int out_size, void* d_ws, size_t ws_size,
                              hipStream_t stream) {
    const float* prev_cont = (const float*)d_in[0];
    const float* F         = (const float*)d_in[1];
    const float* prev_char = (const float*)d_in[2];
    const float* state_h   = (const float*)d_in[3];
    const float* state_c   = (const float*)d_in[4];
    const float* Wc        = (const float*)d_in[5];
    const float* Wct1      = (const float*)d_in[6];
    const float* Wo        = (const float*)d_in[7];
    const float* Wct2      = (const float*)d_in[8];
    const float* lstm_k    = (const float*)d_in[9];
    const float* lstm_r    = (const float*)d_in[10];
    const float* W_h       = (const float*)d_in[11];
    const float* V_attn    = (const float*)d_in[12];

    // d_out layout: O_t [B,V] | c_t [B,CH] | new_h [B,L] | new_c [B,L]
    float* out_Ot = (float*)d_out;
    float* out_ct = out_Ot + (size_t)B * V;
    float* out_h  = out_ct + (size_t)B * CH;
    float* out_c  = out_h  + (size_t)B * L;

    // workspace (floats): lstm_inp [B,E] | attn num [B,CH] | attn den [B,CH]
    float* ws_inp = (float*)d_ws;
    float* ws_num = ws_inp + B * E;
    float* ws_den = ws_num + B * CH;
    const size_t zero_bytes = (size_t)(B * E + 2 * B * CH) * sizeof(float);
    hipMemsetAsync(d_ws, 0, zero_bytes, stream);

    // 1) lstm input GEMM (WMMA, split-K + f32 atomics)
    k_lstm_inp<<<dim3(E / 16, KSPLIT), 32, 0, stream>>>(
        prev_char, Wc, prev_cont, Wct1, ws_inp);

    // 2) LSTM cell -> new_h, clipped new_c directly in d_out
    k_lstm_cell<<<B, L, 0, stream>>>(
        ws_inp, state_h, state_c, lstm_k, lstm_r, out_h, out_c);

    // 3) attention: one streaming pass over F (64 MB), then finalize c_t
    k_attn_partial<<<dim3(B, TCHUNKS), CH, 0, stream>>>(
        out_h, W_h, V_attn, F, ws_num, ws_den);
    k_attn_final<<<B, CH, 0, stream>>>(ws_num, ws_den, out_ct);

    // 4) output logits GEMM (WMMA) into O_t region, then row softmax
    k_logits<<<V / 16, 32, 0, stream>>>(
        out_h, Wo, out_ct, Wct2, out_Ot);
    k_softmax_rows<<<B, 512, 0, stream>>>(out_Ot);
}